// TextBasedEmbedding_57690000720007
// MI455X (gfx1250) — compile-verified
//
#include <hip/hip_runtime.h>

// f32 WMMA operand types (per CDNA5_HIP.md probe-confirmed signatures)
typedef __attribute__((ext_vector_type(2))) float v2f;
typedef __attribute__((ext_vector_type(8))) float v8f;

#define VOCAB     14
#define VOCAB_PAD 16
#define DM        128   // d_model
#define SEGS_PER_BLOCK 16

// One block = 256 threads = 8 waves. Block b handles segments [16b, 16b+16).
// out[t, :] = (counts[t, 0:14] @ emb[0:14, :]) / max(total[t], 1)
// computed with V_WMMA_F32_16X16X4_F32 chained over K=16 (vocab padded).
__global__ __launch_bounds__(256)
void TextBasedEmbedding_segmean_wmma(const float* __restrict__ emb,
                                     const int*   __restrict__ char_idx,
                                     const int*   __restrict__ seg_ids,
                                     float*       __restrict__ out,
                                     int nchars)
{
    __shared__ float s_emb[VOCAB_PAD * DM];          // 16x128, rows 14..15 = 0
    __shared__ float s_cnt[SEGS_PER_BLOCK * VOCAB_PAD]; // 16 segs x 16 vocab (A matrix)
    __shared__ float s_inv[SEGS_PER_BLOCK];          // 1/max(count,1)
    __shared__ int   s_range[2];

    const int tid = threadIdx.x;
    const int s0  = blockIdx.x * SEGS_PER_BLOCK;

    // ---- stage embedding table into LDS (zero-pad rows 14,15), zero counts ----
    for (int i = tid; i < SEGS_PER_BLOCK * VOCAB_PAD; i += 256) s_cnt[i] = 0.0f;
    for (int i = tid; i < 2 * DM; i += 256)                     s_emb[VOCAB * DM + i] = 0.0f;
    for (int i = tid; i < VOCAB * DM; i += 256)                 s_emb[i] = emb[i];

    // ---- segment ids are sorted: binary search char range for [s0, s0+16) ----
    if (tid == 0) {
        int lo = 0, hi = nchars;
        while (lo < hi) { int mid = (lo + hi) >> 1; if (seg_ids[mid] < s0) lo = mid + 1; else hi = mid; }
        s_range[0] = lo;
        int lo2 = lo; hi = nchars;
        while (lo2 < hi) { int mid = (lo2 + hi) >> 1; if (seg_ids[mid] < s0 + SEGS_PER_BLOCK) lo2 = mid + 1; else hi = mid; }
        s_range[1] = lo2;
    }
    __syncthreads();

    // ---- build per-segment vocab histogram in LDS (ds_add_f32) ----
    const int start = s_range[0], end = s_range[1];
    for (int i = start + tid; i < end; i += 256) {
        int s = seg_ids[i] - s0;          // in [0,16) by construction
        int v = char_idx[i];              // in [0,14)
        atomicAdd(&s_cnt[s * VOCAB_PAD + v], 1.0f);
    }
    __syncthreads();

    // ---- per-segment reciprocal counts (reference: / max(count, 1)) ----
    if (tid < SEGS_PER_BLOCK) {
        float t = 0.0f;
#pragma unroll
        for (int v = 0; v < VOCAB; ++v) t += s_cnt[tid * VOCAB_PAD + v];
        s_inv[tid] = 1.0f / fmaxf(t, 1.0f);
    }
    __syncthreads();

    // ---- WMMA: each wave computes one 16x16 tile of out[s0:s0+16, c0:c0+16] ----
    const int wave = tid >> 5;
    const int lane = tid & 31;
    const int nm   = lane & 15;   // M index for A, N index for B/C/D
    const int half = lane >> 4;   // K-pair select for A/B, row-half select for C/D
    const int c0   = wave * 16;   // column tile: 8 waves cover DM=128

    v8f acc = {};
#pragma unroll
    for (int k4 = 0; k4 < 4; ++k4) {
        const int k0 = k4 * 4 + half * 2;  // lanes 0-15: K={k0,k0+1}; lanes 16-31: +2
        v2f a, b;
        // A (16x4 f32): lane nm = row M, VGPR0/1 = consecutive K
        a.x = s_cnt[nm * VOCAB_PAD + k0];
        a.y = s_cnt[nm * VOCAB_PAD + k0 + 1];
        // B (4x16 f32): row K striped across lanes (N = nm)
        b.x = s_emb[k0 * DM + c0 + nm];
        b.y = s_emb[(k0 + 1) * DM + c0 + nm];
        // 8 args: (neg_a, A, neg_b, B, c_mod, C, reuse_a, reuse_b)
        acc = __builtin_amdgcn_wmma_f32_16x16x4_f32(
            false, a, false, b, (short)0, acc, false, false);
    }

    // ---- scale by 1/count and store: C/D VGPR r -> rows r (lanes 0-15) and r+8 ----
#pragma unroll
    for (int r = 0; r < 8; ++r) {
        const int m = r + 8 * half;
        out[(size_t)(s0 + m) * DM + c0 + nm] = acc[r] * s_inv[m];
    }
}

extern "C" void kernel_launch(void* const* d_in, const int* in_sizes, int n_in,
                              void* d_out, int out_size, void* d_ws, size_t ws_size,
                              hipStream_t stream) {
    const float* emb      = (const float*)d_in[0];   // [14, 128] f32
    const int*   char_idx = (const int*)d_in[1];     // [N] int32
    const int*   seg_ids  = (const int*)d_in[2];     // [N] int32, sorted
    float*       out      = (float*)d_out;           // [T, 128] f32

    const int nchars = in_sizes[1];
    const int T      = out_size / DM;                // 131072
    const int blocks = T / SEGS_PER_BLOCK;           // 8192

    hipLaunchKernelGGL(TextBasedEmbedding_segmean_wmma,
                       dim3(blocks), dim3(256), 0, stream,
                       emb, char_idx, seg_ids, out, nchars);
}